// Generator_41618233098576
// MI455X (gfx1250) — compile-verified
//
#include <hip/hip_runtime.h>

typedef __attribute__((ext_vector_type(2))) float v2f;
typedef __attribute__((ext_vector_type(8))) float v8f;

// ---------------------------------------------------------------------------
// Preprocessing: degree (dst-only), dinv = rsqrt(max(deg,1)) or 0, edge norm
// ---------------------------------------------------------------------------
__global__ void deg_kernel(const int* __restrict__ dst, float* __restrict__ deg, int E) {
  int e = blockIdx.x * blockDim.x + threadIdx.x;
  if (e < E) atomicAdd(&deg[dst[e]], 1.0f);
}

__global__ void dinv_kernel(float* __restrict__ deg, int N) {
  int i = blockIdx.x * blockDim.x + threadIdx.x;
  if (i < N) {
    float d = deg[i];
    deg[i] = (d > 0.0f) ? rsqrtf(fmaxf(d, 1.0f)) : 0.0f;  // in-place -> dinv
  }
}

__global__ void norm_kernel(const int* __restrict__ src, const int* __restrict__ dst,
                            const float* __restrict__ dinv, float* __restrict__ norm, int E) {
  int e = blockIdx.x * blockDim.x + threadIdx.x;
  if (e < E) norm[e] = dinv[src[e]] * dinv[dst[e]];
}

// ---------------------------------------------------------------------------
// Repack one layer's W[4,128,128] (row-major) into WMMA-B-fragment order:
//   Wr[mat][kq][hi][col] = { W[mat][kq*4+2*hi][col], W[mat][kq*4+2*hi+1][col] }
// pair index i: col = i&127, hi = (i>>7)&1, kq = (i>>8)&31, mat = i>>13
// total pairs per layer = 4*32*2*128 = 32768
// ---------------------------------------------------------------------------
__global__ void repack_w_kernel(const float* __restrict__ W, float* __restrict__ Wr) {
  int i = blockIdx.x * blockDim.x + threadIdx.x;
  if (i >= 32768) return;
  int col = i & 127;
  int hi  = (i >> 7) & 1;
  int kq  = (i >> 8) & 31;
  int mat = i >> 13;
  int k = kq * 4 + hi * 2;
  const float* w = W + (size_t)mat * 16384;
  v2f p;
  p.x = w[(size_t)k * 128 + col];
  p.y = w[(size_t)(k + 1) * 128 + col];
  ((v2f*)Wr)[i] = p;
}

// ---------------------------------------------------------------------------
// One hop of normalized aggregation: out[dst] += norm * in[src]
// 32 threads per edge, 4 feature dims each (float4 gather + 4 f32 atomics)
// ---------------------------------------------------------------------------
__global__ void scatter_kernel(const float* __restrict__ in, float* __restrict__ out,
                               const int* __restrict__ src, const int* __restrict__ dst,
                               const float* __restrict__ norm, long total) {
  long t = (long)blockIdx.x * blockDim.x + threadIdx.x;
  if (t >= total) return;
  int e = (int)(t >> 5);
  int q = (int)(t & 31);
  int s = src[e];
  int d = dst[e];
  float w = norm[e];
  const float4 v = *(const float4*)(in + (size_t)s * 128 + q * 4);
  float* p = out + (size_t)d * 128 + q * 4;
  atomicAdd(p + 0, w * v.x);
  atomicAdd(p + 1, w * v.y);
  atomicAdd(p + 2, w * v.z);
  atomicAdd(p + 3, w * v.w);
}

// ---------------------------------------------------------------------------
// C[M,128] (+)= A[M,128] @ W[128,128] via V_WMMA_F32_16X16X4_F32.
// Repacked W (64KB) staged into LDS per block; each wave computes a 32x128
// output stripe (16 v8f accumulators). Both operands are software-pipelined
// one k-quad ahead, and __builtin_amdgcn_sched_barrier(0) fences the loads
// from the WMMA burst so the backend cannot sink them back (keeps the
// dscnt/loadcnt waits between bursts, not inside them).
// ---------------------------------------------------------------------------
__global__ void __launch_bounds__(256)
tag_gemm_kernel(const float* __restrict__ A, const float* __restrict__ Wr,
                float* __restrict__ C, int M, int accum) {
  __shared__ float lw[128 * 128];  // 64 KB repacked weights

  // cooperative LDS fill (all waves participate before any early exit)
  {
    const float4* s4 = (const float4*)Wr;
    float4* d4 = (float4*)lw;
    for (int i = threadIdx.x; i < 128 * 128 / 4; i += 256) d4[i] = s4[i];
  }
  __syncthreads();

  int wg     = blockIdx.x * 8 + (threadIdx.x >> 5);
  int mtiles = M >> 4;
  int t0 = wg * 2;
  int t1 = wg * 2 + 1;
  if (t0 >= mtiles) return;
  bool has1 = (t1 < mtiles);

  int lane   = threadIdx.x & 31;
  int lane16 = lane & 15;
  int hi     = lane >> 4;  // 0 or 1
  int mb0 = t0 << 4;
  int mb1 = (has1 ? t1 : t0) << 4;  // clamp tail: duplicate compute, guarded store

  v8f acc[16];
  if (accum) {
#pragma unroll
    for (int nt = 0; nt < 8; ++nt)
#pragma unroll
      for (int r = 0; r < 8; ++r) {
        acc[nt][r]     = C[(size_t)(mb0 + r + hi * 8) * 128 + nt * 16 + lane16];
        acc[8 + nt][r] = C[(size_t)(mb1 + r + hi * 8) * 128 + nt * 16 + lane16];
      }
  } else {
#pragma unroll
    for (int nt = 0; nt < 16; ++nt)
#pragma unroll
      for (int r = 0; r < 8; ++r) acc[nt][r] = 0.0f;
  }

  const float* arow0 = A + (size_t)(mb0 + lane16) * 128 + hi * 2;
  const float* arow1 = A + (size_t)(mb1 + lane16) * 128 + hi * 2;
  const v2f* bbase = (const v2f*)lw + hi * 128 + lane16;

  // prologue: operands for k-quad 0
  v2f bcur[8], bnext[8];
#pragma unroll
  for (int nt = 0; nt < 8; ++nt) bcur[nt] = bbase[nt * 16];
  v2f a0c = *(const v2f*)(arow0);
  v2f a1c = *(const v2f*)(arow1);
  v2f a0n, a1n;

#pragma unroll
  for (int kq = 0; kq < 32; ++kq) {
    // prefetch next k-quad's operands (issued before the burst, used after)
    if (kq + 1 < 32) {
      const v2f* bq = bbase + (kq + 1) * 256;
#pragma unroll
      for (int nt = 0; nt < 8; ++nt) bnext[nt] = bq[nt * 16];
      a0n = *(const v2f*)(arow0 + (kq + 1) * 4);
      a1n = *(const v2f*)(arow1 + (kq + 1) * 4);
    }
    __builtin_amdgcn_sched_barrier(0);

#pragma unroll
    for (int nt = 0; nt < 8; ++nt) {
      acc[nt] = __builtin_amdgcn_wmma_f32_16x16x4_f32(
          false, a0c, false, bcur[nt], (short)0, acc[nt], false, false);
      acc[8 + nt] = __builtin_amdgcn_wmma_f32_16x16x4_f32(
          false, a1c, false, bcur[nt], (short)0, acc[8 + nt], false, false);
    }
    __builtin_amdgcn_sched_barrier(0);

    if (kq + 1 < 32) {
#pragma unroll
      for (int nt = 0; nt < 8; ++nt) bcur[nt] = bnext[nt];  // register rotation
      a0c = a0n;
      a1c = a1n;
    }
  }

#pragma unroll
  for (int nt = 0; nt < 8; ++nt)
#pragma unroll
    for (int r = 0; r < 8; ++r)
      C[(size_t)(mb0 + r + hi * 8) * 128 + nt * 16 + lane16] = acc[nt][r];
  if (has1) {
#pragma unroll
    for (int nt = 0; nt < 8; ++nt)
#pragma unroll
      for (int r = 0; r < 8; ++r)
        C[(size_t)(mb1 + r + hi * 8) * 128 + nt * 16 + lane16] = acc[8 + nt][r];
  }
}

// ---------------------------------------------------------------------------
// out = prelu(in + bias, a)
// ---------------------------------------------------------------------------
__global__ void bias_prelu_kernel(const float* __restrict__ in, const float* __restrict__ b,
                                  const float* __restrict__ a, float* __restrict__ out,
                                  long total) {
  long i = (long)blockIdx.x * blockDim.x + threadIdx.x;
  if (i >= total) return;
  float v = in[i] + b[i & 127];
  out[i] = (v >= 0.0f) ? v : a[0] * v;
}

// ---------------------------------------------------------------------------
extern "C" void kernel_launch(void* const* d_in, const int* in_sizes, int n_in,
                              void* d_out, int out_size, void* d_ws, size_t ws_size,
                              hipStream_t stream) {
  const float* x  = (const float*)d_in[0];
  const int*   ei = (const int*)d_in[1];
  const float* Wl[3] = {(const float*)d_in[2], (const float*)d_in[5], (const float*)d_in[8]};
  const float* bl[3] = {(const float*)d_in[3], (const float*)d_in[6], (const float*)d_in[9]};
  const float* al[3] = {(const float*)d_in[4], (const float*)d_in[7], (const float*)d_in[10]};

  const int N = in_sizes[0] / 128;
  const int E = in_sizes[1] / 2;
  const int* src = ei;
  const int* dst = ei + E;

  // workspace layout (floats)
  float* dinv = (float*)d_ws;                       // N   (deg -> dinv in place)
  float* norm = dinv + N;                           // E
  float* bufA = norm + E;                           // N*128
  float* bufB = bufA + (size_t)N * 128;             // N*128
  float* outb = bufB + (size_t)N * 128;             // N*128
  float* wrep = outb + (size_t)N * 128;             // 3 * 4 * 128*128 repacked weights
  const size_t featBytes = (size_t)N * 128 * sizeof(float);

  const int TPB = 256;
  const int gE  = (E + TPB - 1) / TPB;
  const int gN  = (N + TPB - 1) / TPB;
  const long scatterTotal = (long)E * 32;
  const int  gScatter     = (int)((scatterTotal + TPB - 1) / TPB);
  const int  mtiles       = N / 16;                       // 50000 -> 3125 exact
  const int  gWaves       = (mtiles + 1) / 2;             // 32 rows per wave
  const int  gGemm        = (gWaves + 7) / 8;             // 8 waves / block
  const long elemTotal    = (long)N * 128;
  const int  gElem        = (int)((elemTotal + TPB - 1) / TPB);

  // --- preprocessing: deg -> dinv -> norm, weight repack ---
  hipMemsetAsync(dinv, 0, (size_t)N * sizeof(float), stream);
  deg_kernel<<<gE, TPB, 0, stream>>>(dst, dinv, E);
  dinv_kernel<<<gN, TPB, 0, stream>>>(dinv, N);
  norm_kernel<<<gE, TPB, 0, stream>>>(src, dst, dinv, norm, E);
  for (int l = 0; l < 3; ++l)
    repack_w_kernel<<<128, TPB, 0, stream>>>(Wl[l], wrep + (size_t)l * 4 * 16384);

  // --- 3 TAGConv layers ---
  const float* curIn = x;
  for (int l = 0; l < 3; ++l) {
    const float* wr = wrep + (size_t)l * 4 * 16384;
    // out = h0 @ W[0]
    tag_gemm_kernel<<<gGemm, TPB, 0, stream>>>(curIn, wr, outb, N, 0);

    float* targ[3] = {bufA, bufB, bufA};
    const float* hp = curIn;
    for (int k = 1; k <= 3; ++k) {
      float* t = targ[k - 1];
      hipMemsetAsync(t, 0, featBytes, stream);
      scatter_kernel<<<gScatter, TPB, 0, stream>>>(hp, t, src, dst, norm, scatterTotal);
      tag_gemm_kernel<<<gGemm, TPB, 0, stream>>>(t, wr + (size_t)k * 16384, outb, N, 1);
      hp = t;
    }

    float* lout = (l == 2) ? (float*)d_out : bufB;
    bias_prelu_kernel<<<gElem, TPB, 0, stream>>>(outb, bl[l], al[l], lout, elemTotal);
    curIn = lout;
  }
}